// LocalHierarchicalKLLoss_28011776704809
// MI455X (gfx1250) — compile-verified
//
#include <hip/hip_runtime.h>
#include <hip/hip_bf16.h>
#include <math.h>

#define N_DOCS   256
#define L_TOK    8192
#define S_SENT   128
#define K_OFFSET 1
#define ALPHA_C  0.05f
#define EPS_C    1e-30f

#define THREADS    256
#define PER_THREAD (L_TOK / THREADS)   // 32 tokens per thread
#define VEC_ITERS  (PER_THREAD / 4)    // 8 x float4/int4 per thread

struct __align__(16) SMem {
    float         attn[L_TOK];    // 32 KB, filled by async DMA
    unsigned char mask[L_TOK];    // 8 KB evidence-token mask
    float         red[32];        // 4 reduction slots x 8 waves
    int           flag;           // has-evidence flag
};

__device__ __forceinline__ float waveReduceSum(float v) {
#pragma unroll
    for (int off = 16; off > 0; off >>= 1)
        v += __shfl_xor(v, off, 32);   // wave32 on gfx1250
    return v;
}

__global__ __launch_bounds__(THREADS)
void klloss_main(const float* __restrict__ doc_attn,
                 const int*   __restrict__ sent_labels,
                 const int*   __restrict__ token_labels,
                 const int*   __restrict__ sent_pos,
                 float*       __restrict__ ws_loss,
                 int*         __restrict__ ws_flag)
{
    __shared__ SMem sm;
    const int tid = threadIdx.x;
    const int row = blockIdx.x;

    const int* trow = token_labels + (size_t)row * L_TOK;

    // ---- Phase 0a: async global->LDS DMA of this row's attention ----------
    // 256 lanes x 16B x 8 iters = 32 KB via the ASYNC engine; overlaps with
    // the prefetches, mask zeroing and mask construction below.
    {
        const float* gsrc = doc_attn + (size_t)row * L_TOK;
        unsigned int lds_base = (unsigned int)(uintptr_t)(&sm.attn[0]);
#pragma unroll
        for (int it = 0; it < VEC_ITERS; ++it) {
            unsigned int elem = (unsigned int)(it * (THREADS * 4) + tid * 4);
            unsigned int ldsa = lds_base + elem * 4u;
            const float* g = gsrc + elem;
            asm volatile("global_load_async_to_lds_b128 %0, %1, off"
                         :: "v"(ldsa), "v"(g) : "memory");
        }
    }

    // ---- Phase 0b: prefetch the token-label row (global_prefetch_b8) ------
    // One byte per thread, 128B-cacheline stride: 256 lines == full 32 KB row
    // pulled toward cache while the DMA and mask build proceed.
    __builtin_prefetch(&trow[tid * 32], 0, 1);
    // Warm the sentence metadata used in Phase 1 (1.5 KB total per row).
    if (tid < S_SENT) {
        __builtin_prefetch(&sent_labels[row * S_SENT + tid], 0, 1);
        __builtin_prefetch(&sent_pos[(row * S_SENT + tid) * 2], 0, 1);
    }

    // ---- Zero mask + flag while the DMA is in flight ----------------------
    {
        uint4 z = make_uint4(0u, 0u, 0u, 0u);
        uint4* mw = (uint4*)&sm.mask[0];     // 512 uint4 total
        mw[tid]           = z;
        mw[tid + THREADS] = z;
        if (tid == 0) sm.flag = 0;
    }
    __syncthreads();

    // ---- Phase 1: mark tokens covered by evidence sentences ---------------
    if (tid < S_SENT) {
        const int sidx = row * S_SENT + tid;
        const int sl = sent_labels[sidx];
        if (sl > 0) {
            sm.flag = 1;                      // idempotent, race-free
            int st = sent_pos[sidx * 2 + 0] + K_OFFSET;
            int en = sent_pos[sidx * 2 + 1] + K_OFFSET;
            if (st < 0) st = 0;
            if (en > L_TOK) en = L_TOK;
            for (int t = st; t < en; ++t) sm.mask[t] = 1;
        }
    }

    // DMA must have landed before sm.attn is read; the barrier then makes
    // every wave's async-copied bytes + the mask visible workgroup-wide.
    asm volatile("s_wait_asynccnt 0" ::: "memory");
    __syncthreads();

    // ---- Pass 1: n_tok, sum(max(a,EPS)*m), sum(tl) ------------------------
    float4 sv[VEC_ITERS];          // clamped scores (kept in VGPRs)
    float4 tlv[VEC_ITERS];         // token_label * mask
    unsigned int mword[VEC_ITERS]; // 4 mask bytes per word

    float accN = 0.f, accS = 0.f, accT = 0.f;

#pragma unroll
    for (int it = 0; it < VEC_ITERS; ++it) {
        const int elem = it * (THREADS * 4) + tid * 4;   // coalesced layout
        float4 a  = *(const float4*)&sm.attn[elem];      // ds_load_b128
        int4   tk = *(const int4*)&trow[elem];           // global_load_b128
        unsigned int mw = *(const unsigned int*)&sm.mask[elem];
        mword[it] = mw;

        float m0 = (float)((mw      ) & 1u);
        float m1 = (float)((mw >>  8) & 1u);
        float m2 = (float)((mw >> 16) & 1u);
        float m3 = (float)((mw >> 24) & 1u);

        float s0 = fmaxf(a.x, EPS_C), s1 = fmaxf(a.y, EPS_C);
        float s2 = fmaxf(a.z, EPS_C), s3 = fmaxf(a.w, EPS_C);
        sv[it] = make_float4(s0, s1, s2, s3);

        float t0 = (float)tk.x * m0, t1 = (float)tk.y * m1;
        float t2 = (float)tk.z * m2, t3 = (float)tk.w * m3;
        tlv[it] = make_float4(t0, t1, t2, t3);

        accN += (m0 + m1) + (m2 + m3);
        accS += (s0 * m0 + s1 * m1) + (s2 * m2 + s3 * m3);
        accT += (t0 + t1) + (t2 + t3);
    }

    accN = waveReduceSum(accN);
    accS = waveReduceSum(accS);
    accT = waveReduceSum(accT);

    const int wid = tid >> 5, lane = tid & 31;
    if (lane == 0) {
        sm.red[0 * 8 + wid] = accN;
        sm.red[1 * 8 + wid] = accS;
        sm.red[2 * 8 + wid] = accT;
    }
    __syncthreads();

    float nTok = 0.f, Ssum = 0.f, Tsum = 0.f;
#pragma unroll
    for (int w = 0; w < 8; ++w) {
        nTok += sm.red[0 * 8 + w];
        Ssum += sm.red[1 * 8 + w];
        Tsum += sm.red[2 * 8 + w];
    }

    // ---- Pass 2: KL terms over masked tokens (branchless, from VGPRs) -----
    const float sden = Ssum + EPS_C;
    const float tden = Tsum + EPS_C;
    const float un   = ALPHA_C * (1.0f / fmaxf(nTok, 1.0f)); // ALPHA*uniform

    float acc = 0.f;
#pragma unroll
    for (int it = 0; it < VEC_ITERS; ++it) {
        const unsigned int mw = mword[it];
        const float4 s  = sv[it];
        const float4 tl = tlv[it];
        {
            float m = (float)((mw      ) & 1u);
            float q = (1.0f - ALPHA_C) * (tl.x / tden) + un;
            float lp = logf(s.x / sden + EPS_C);
            acc += m * (q * (logf(q) - lp));
        }
        {
            float m = (float)((mw >>  8) & 1u);
            float q = (1.0f - ALPHA_C) * (tl.y / tden) + un;
            float lp = logf(s.y / sden + EPS_C);
            acc += m * (q * (logf(q) - lp));
        }
        {
            float m = (float)((mw >> 16) & 1u);
            float q = (1.0f - ALPHA_C) * (tl.z / tden) + un;
            float lp = logf(s.z / sden + EPS_C);
            acc += m * (q * (logf(q) - lp));
        }
        {
            float m = (float)((mw >> 24) & 1u);
            float q = (1.0f - ALPHA_C) * (tl.w / tden) + un;
            float lp = logf(s.w / sden + EPS_C);
            acc += m * (q * (logf(q) - lp));
        }
    }

    acc = waveReduceSum(acc);
    if (lane == 0) sm.red[3 * 8 + wid] = acc;   // slot 3: no WAR on slots 0-2
    __syncthreads();

    if (tid == 0) {
        float loss = 0.f;
#pragma unroll
        for (int w = 0; w < 8; ++w) loss += sm.red[3 * 8 + w];
        ws_loss[row] = loss;
        ws_flag[row] = sm.flag;
    }
}

__global__ __launch_bounds__(THREADS)
void klloss_final(const float* __restrict__ ws_loss,
                  const int*   __restrict__ ws_flag,
                  float*       __restrict__ out)
{
    __shared__ float sred[8];
    __shared__ float scnt[8];
    const int tid = threadIdx.x;

    float v = 0.f, c = 0.f;
    if (tid < N_DOCS) {
        if (ws_flag[tid] != 0) { v = ws_loss[tid]; c = 1.f; }
    }
    v = waveReduceSum(v);
    c = waveReduceSum(c);

    const int wid = tid >> 5, lane = tid & 31;
    if (lane == 0) { sred[wid] = v; scnt[wid] = c; }
    __syncthreads();

    if (tid == 0) {
        float tot = 0.f, cnt = 0.f;
#pragma unroll
        for (int w = 0; w < 8; ++w) { tot += sred[w]; cnt += scnt[w]; }
        out[0] = tot / fmaxf(cnt, 1.0f);
    }
}

extern "C" void kernel_launch(void* const* d_in, const int* in_sizes, int n_in,
                              void* d_out, int out_size, void* d_ws, size_t ws_size,
                              hipStream_t stream) {
    (void)in_sizes; (void)n_in; (void)out_size; (void)ws_size;
    const float* doc_attn     = (const float*)d_in[0];
    const int*   sent_labels  = (const int*)d_in[1];
    const int*   token_labels = (const int*)d_in[2];
    const int*   sent_pos     = (const int*)d_in[3];

    float* ws_loss = (float*)d_ws;
    int*   ws_flag = (int*)((char*)d_ws + N_DOCS * sizeof(float));
    float* out     = (float*)d_out;

    klloss_main<<<dim3(N_DOCS), dim3(THREADS), 0, stream>>>(
        doc_attn, sent_labels, token_labels, sent_pos, ws_loss, ws_flag);
    klloss_final<<<dim3(1), dim3(THREADS), 0, stream>>>(ws_loss, ws_flag, out);
}